// VertexUpdate_91096256348930
// MI455X (gfx1250) — compile-verified
//
#include <hip/hip_runtime.h>

typedef __attribute__((ext_vector_type(2))) float v2f;
typedef __attribute__((ext_vector_type(8))) float v8f;

#define D_E   32
#define D_V   64
#define D_G   64
#define HDIM  256
#define D_OUT 64
#define NIN   256     // D_V + 4*D_E + D_G
#define MT    64      // vertices (rows) per MLP block
#define NTHR  512     // 16 waves
#define LDP   260     // padded LDS row stride (floats)

// ---- order-preserving float <-> uint mapping for atomic min/max ----
__device__ __forceinline__ unsigned mapf(float f) {
  unsigned u = __float_as_uint(f);
  return (u & 0x80000000u) ? ~u : (u | 0x80000000u);
}
__device__ __forceinline__ float unmapf(unsigned u) {
  unsigned s = (u & 0x80000000u) ? (u ^ 0x80000000u) : ~u;
  return __uint_as_float(s);
}

// ---- workspace init: sum=0, cnt=0, min=+inf(mapped)=0xFFFFFFFF, max=0 ----
__global__ void init_ws(float* sumb, float* cntb, unsigned* minb, unsigned* maxb, int nv) {
  int tid = blockIdx.x * blockDim.x + threadIdx.x;
  int total = nv * D_E;
  if (tid < total) {
    sumb[tid] = 0.0f;
    minb[tid] = 0xFFFFFFFFu;
    maxb[tid] = 0u;
    if (tid < nv) cntb[tid] = 0.0f;
  }
}

// ---- edge scatter: one thread per (edge, feature); wave == one edge row ----
__global__ void edge_scatter(const int* __restrict__ src, const float* __restrict__ e_attr,
                             float* sumb, float* cntb, unsigned* minb, unsigned* maxb,
                             int ne) {
  long long tid = (long long)blockIdx.x * blockDim.x + threadIdx.x;
  int e = (int)(tid >> 5);
  int f = (int)(tid & 31);
  if (e >= ne) return;
  int s = src[e];                              // wave-uniform
  float v = e_attr[(long long)e * D_E + f];    // coalesced 128B per wave
  int o = s * D_E + f;
  atomicAdd(&sumb[o], v);
  unsigned m = mapf(v);
  atomicMin(&minb[o], m);
  atomicMax(&maxb[o], m);
  if (f == 0) atomicAdd(&cntb[s], 1.0f);
}

// ---- one dense layer via fp32 WMMA 16x16x4 ----
// hin: [MT][LDP] LDS, K = NIN. W row-major [NIN][wstride] in global (L2-hot).
// 16 waves: row-tile = wave&3, column-tiles = (wave>>2) + 4*j, j<NTW.
// A fragments for a 32-deep K chunk are held in registers and reused across
// the NTW column tiles; NTW independent accumulator chains keep the matrix
// pipe busy between B loads.
template <int NTW>
__device__ __forceinline__ void layer_wmma(const float* __restrict__ hin,
                                           float* __restrict__ hout,
                                           const float* __restrict__ W,
                                           const float* __restrict__ b,
                                           int wstride, bool do_relu,
                                           int wave, int lane) {
  int lr   = lane & 15;   // A row within half / B,C,D column within tile
  int half = lane >> 4;   // 0: K={0,1}; 1: K={2,3}
  int rt   = wave & 3;
  int col0 = (wave >> 2) * 16 + lr;

  v8f acc[NTW];
#pragma unroll
  for (int j = 0; j < NTW; ++j) {
    float bv = b[col0 + 64 * j];
    acc[j] = {bv, bv, bv, bv, bv, bv, bv, bv};
  }

  const float* arow = &hin[(rt * 16 + lr) * LDP];
  for (int kc = 0; kc < NIN; kc += 32) {   // 8 WMMA k-steps per chunk
    v2f afrag[8];
#pragma unroll
    for (int k = 0; k < 8; ++k) {
      int kk = kc + 4 * k + 2 * half;
      afrag[k].x = arow[kk];
      afrag[k].y = arow[kk + 1];
    }
#pragma unroll
    for (int k = 0; k < 8; ++k) {
      int kk = kc + 4 * k + 2 * half;
      const float* wr0 = &W[(long long)kk * wstride + col0];
      const float* wr1 = wr0 + wstride;
#pragma unroll
      for (int j = 0; j < NTW; ++j) {
        v2f bb;
        bb.x = wr0[64 * j];   // immediate-offset loads, 256B apart
        bb.y = wr1[64 * j];
        acc[j] = __builtin_amdgcn_wmma_f32_16x16x4_f32(
            false, afrag[k], false, bb, (short)0, acc[j], false, false);
      }
    }
  }

#pragma unroll
  for (int j = 0; j < NTW; ++j) {
#pragma unroll
    for (int i = 0; i < 8; ++i) {
      float x = acc[j][i];
      if (do_relu) x = fmaxf(x, 0.0f);
      // C/D layout: VGPR i -> M = i (lanes 0-15), M = i+8 (lanes 16-31)
      hout[(rt * 16 + i + 8 * half) * LDP + col0 + 64 * j] = x;
    }
  }
}

__global__ void __launch_bounds__(NTHR)
mlp_fused(const float* __restrict__ v_attr, const float* __restrict__ g,
          const int* __restrict__ batch,
          const float* __restrict__ W1, const float* __restrict__ b1,
          const float* __restrict__ W2, const float* __restrict__ b2,
          const float* __restrict__ W3, const float* __restrict__ b3,
          const float* __restrict__ sumb, const float* __restrict__ cntb,
          const unsigned* __restrict__ minb, const unsigned* __restrict__ maxb,
          float* __restrict__ out, int nv) {
  extern __shared__ float smem[];
  float* buf0 = smem;             // [MT][LDP]
  float* buf1 = smem + MT * LDP;  // [MT][LDP]
  int tid  = threadIdx.x;
  int wave = tid >> 5;
  int lane = tid & 31;
  int base = blockIdx.x * MT;

  // stage concatenated input: [v_attr(64) | min(32) | mean(32) | sum(32) | max(32) | g(64)]
  for (int idx = tid; idx < MT * NIN; idx += NTHR) {
    int r = idx >> 8;       // NIN == 256
    int c = idx & 255;
    int v = base + r;
    float x = 0.0f;
    if (v < nv) {
      if (c < 64) {
        x = v_attr[(long long)v * D_V + c];
      } else if (c < 192) {
        int f = c & 31;
        int o = v * D_E + f;
        float cnt = cntb[v];
        if (c < 96)       x = (cnt > 0.0f) ? unmapf(minb[o]) : 0.0f;   // min
        else if (c < 128) x = sumb[o] / fmaxf(cnt, 1.0f);               // mean
        else if (c < 160) x = sumb[o];                                  // sum
        else              x = (cnt > 0.0f) ? unmapf(maxb[o]) : 0.0f;    // max
      } else {
        x = g[batch[v] * D_G + (c - 192)];
      }
    }
    buf0[r * LDP + c] = x;
  }
  __syncthreads();

  layer_wmma<4>(buf0, buf1, W1, b1, /*wstride=*/HDIM,  /*relu=*/true,  wave, lane);
  __syncthreads();
  layer_wmma<4>(buf1, buf0, W2, b2, /*wstride=*/HDIM,  /*relu=*/true,  wave, lane);
  __syncthreads();
  layer_wmma<1>(buf0, buf1, W3, b3, /*wstride=*/D_OUT, /*relu=*/false, wave, lane);
  __syncthreads();

  for (int idx = tid; idx < MT * D_OUT; idx += NTHR) {
    int r = idx >> 6;
    int c = idx & 63;
    int v = base + r;
    if (v < nv) out[(long long)v * D_OUT + c] = buf1[r * LDP + c];
  }
}

extern "C" void kernel_launch(void* const* d_in, const int* in_sizes, int n_in,
                              void* d_out, int out_size, void* d_ws, size_t ws_size,
                              hipStream_t stream) {
  const float* v_attr = (const float*)d_in[0];
  const int*   edge   = (const int*)d_in[1];   // row 0 = src vertex per edge
  const float* e_attr = (const float*)d_in[2];
  const float* g      = (const float*)d_in[3];
  const int*   batch  = (const int*)d_in[4];
  const float* W1     = (const float*)d_in[5];
  const float* b1     = (const float*)d_in[6];
  const float* W2     = (const float*)d_in[7];
  const float* b2     = (const float*)d_in[8];
  const float* W3     = (const float*)d_in[9];
  const float* b3     = (const float*)d_in[10];
  float* out = (float*)d_out;

  int nv = in_sizes[0] / D_V;
  int ne = in_sizes[2] / D_E;

  // workspace: sum[nv*32] f32 | min[nv*32] u32 | max[nv*32] u32 | cnt[nv] f32
  float*    sumb = (float*)d_ws;
  unsigned* minb = (unsigned*)(sumb + (size_t)nv * D_E);
  unsigned* maxb = minb + (size_t)nv * D_E;
  float*    cntb = (float*)(maxb + (size_t)nv * D_E);

  int t1 = nv * D_E;
  init_ws<<<(t1 + 255) / 256, 256, 0, stream>>>(sumb, cntb, minb, maxb, nv);

  long long te = (long long)ne * D_E;
  edge_scatter<<<(int)((te + 255) / 256), 256, 0, stream>>>(
      edge, e_attr, sumb, cntb, minb, maxb, ne);

  int nblk = (nv + MT - 1) / MT;
  size_t sh = (size_t)2 * MT * LDP * sizeof(float);  // ~133 KB LDS
  mlp_fused<<<nblk, NTHR, sh, stream>>>(v_attr, g, batch,
                                        W1, b1, W2, b2, W3, b3,
                                        sumb, cntb, minb, maxb, out, nv);
}